// GATNetwork_15487652069902
// MI455X (gfx1250) — compile-verified
//
#include <hip/hip_runtime.h>
#include <math.h>

#define HEADS 4
#define HC 128            // HEADS * C_OUT
#define NEG_SLOPE 0.2f
#define BN_EPS 1e-5f

typedef __attribute__((ext_vector_type(2))) float v2f;
typedef __attribute__((ext_vector_type(8))) float v8f;

// ---- monotonic float<->u32 mapping for atomic float-max via atomicMax(u32) ----
__device__ __forceinline__ unsigned floatFlip(float f) {
  unsigned u = __float_as_uint(f);
  return (u & 0x80000000u) ? ~u : (u | 0x80000000u);
}
__device__ __forceinline__ float floatUnflip(unsigned u) {
  u = (u & 0x80000000u) ? (u & 0x7fffffffu) : ~u;
  return __uint_as_float(u);
}
// floatFlip(-inf) == 0x007FFFFF : init value for the max buffers
#define NEG_INF_FLIPPED 0x007FFFFFu

// ------------------------------ fills ------------------------------
__global__ void k_fill_f32(float* __restrict__ p, float v, size_t n) {
  size_t i = (size_t)blockIdx.x * blockDim.x + threadIdx.x;
  if (i < n) p[i] = v;
}
__global__ void k_fill_u32(unsigned* __restrict__ p, unsigned v, size_t n) {
  size_t i = (size_t)blockIdx.x * blockDim.x + threadIdx.x;
  if (i < n) p[i] = v;
}

// ---------------- self-loop attr: per-dst degree & weight sum ----------------
__global__ void k_degsum(const int* __restrict__ dst, const float* __restrict__ w,
                         float* __restrict__ deg, float* __restrict__ sumw, int nE) {
  int e = blockIdx.x * blockDim.x + threadIdx.x;
  if (e >= nE) return;
  int d = dst[e];
  atomicAdd(&deg[d], 1.0f);
  atomicAdd(&sumw[d], w[e]);
}
__global__ void k_loopattr(const float* __restrict__ deg, const float* __restrict__ sumw,
                           float* __restrict__ loopat, int n) {
  int i = blockIdx.x * blockDim.x + threadIdx.x;
  if (i < n) loopat[i] = sumw[i] / fmaxf(deg[i], 1.0f);
}

// ---------------- f32 WMMA GEMM: out[N,128] = X[N,128] @ W[128,128] + b ----------------
// One wave32 computes one 16x16 output tile; K loop of 32 x v_wmma_f32_16x16x4_f32.
// A 16x4 f32 frag: lanes 0-15 hold row M=lane K={k,k+1}; lanes 16-31 K={k+2,k+3}.
// B 4x16 f32 frag mirrors A: VGPR0 row K=k (lanes<16) / k+2 (lanes>=16), VGPR1 k+1 / k+3.
// C/D 16x16: VGPR j -> M=j (lanes<16) / j+8 (lanes>=16), N=lane&15.
__global__ void __launch_bounds__(256) k_gemm_wmma(const float* __restrict__ X,
                                                   const float* __restrict__ W,
                                                   const float* __restrict__ bias,
                                                   float* __restrict__ out, int nrows) {
  int wave = (blockIdx.x * blockDim.x + threadIdx.x) >> 5;
  int lane = threadIdx.x & 31;
  int colTile = wave & 7;        // 128/16 = 8 column tiles
  int rowTile = wave >> 3;
  int row0 = rowTile * 16;
  if (row0 >= nrows) return;
  int col0 = colTile * 16;
  int half = lane >> 4;          // 0: K lo pair, 1: K hi pair
  int l16  = lane & 15;

  const float* xrow = X + (size_t)(row0 + l16) * HC;
  const float* wcol = W + (col0 + l16);

  v8f acc = {};
#pragma unroll
  for (int k = 0; k < HC; k += 4) {
    int kk = k + 2 * half;
    v2f a, b;
    a.x = xrow[kk];
    a.y = xrow[kk + 1];
    b.x = wcol[(size_t)kk * HC];
    b.y = wcol[(size_t)(kk + 1) * HC];
    acc = __builtin_amdgcn_wmma_f32_16x16x4_f32(false, a, false, b, (short)0, acc,
                                                false, false);
  }
  float bv = bias[col0 + l16];
#pragma unroll
  for (int j = 0; j < 8; ++j) {
    int row = row0 + j + 8 * half;
    out[(size_t)row * HC + col0 + l16] = acc[j] + bv;
  }
}

// ---------------- edge pass 1: attention logits + segment max ----------------
// One wave32 per edge; lane = channel-within-head. Wave-reduce 32-ch dot per head.
__global__ void k_logits(const int* __restrict__ src, const int* __restrict__ dst,
                         const float* __restrict__ ew, const float* __restrict__ loopat,
                         const float* __restrict__ xl, const float* __restrict__ xr,
                         const float* __restrict__ We, const float* __restrict__ att,
                         float* __restrict__ elog, unsigned* __restrict__ mbuf,
                         int nE, int nN) {
  int wid  = (int)(((size_t)blockIdx.x * blockDim.x + threadIdx.x) >> 5);
  int lane = threadIdx.x & 31;
  int tot = nE + nN;
  if (wid >= tot) return;
  int s, d; float wgt;
  if (wid < nE) { s = src[wid]; d = dst[wid]; wgt = ew[wid]; }
  else          { s = d = wid - nE;           wgt = loopat[s]; }
  const float* xls = xl + (size_t)s * HC;
  const float* xrd = xr + (size_t)d * HC;
#pragma unroll
  for (int h = 0; h < HEADS; ++h) {
    int c = h * 32 + lane;
    float v = xls[c] + xrd[c] + wgt * We[c];
    v = v > 0.f ? v : NEG_SLOPE * v;           // leaky_relu
    v *= att[c];                               // att[h][lane]
#pragma unroll
    for (int off = 16; off; off >>= 1) v += __shfl_xor(v, off, 32);
    if (lane == 0) {
      elog[(size_t)wid * HEADS + h] = v;
      atomicMax(&mbuf[(size_t)d * HEADS + h], floatFlip(v));
    }
  }
}

// ---------------- edge pass 2: exp(logit - max) + segment sum ----------------
__global__ void k_expsum(const int* __restrict__ dst, float* __restrict__ elog,
                         const unsigned* __restrict__ mbuf, float* __restrict__ sbuf,
                         int nE, int nN) {
  int idx = blockIdx.x * blockDim.x + threadIdx.x;
  int tot = (nE + nN) * HEADS;
  if (idx >= tot) return;
  int e = idx >> 2, h = idx & 3;
  int d = (e < nE) ? dst[e] : (e - nE);
  float mm = floatUnflip(mbuf[(size_t)d * HEADS + h]);
  if (!isfinite(mm)) mm = 0.f;                 // reference's empty-segment guard
  float ex = __expf(elog[idx] - mm);
  elog[idx] = ex;
  atomicAdd(&sbuf[(size_t)d * HEADS + h], ex);
}

// ---------------- edge pass 3: normalize + weighted scatter-add ----------------
// One wave32 per edge; lane covers channels {lane, lane+32, lane+64, lane+96}
// => channel block j == head j; accesses are 128B-contiguous per wave.
__global__ void k_scatter(const int* __restrict__ src, const int* __restrict__ dst,
                          const float* __restrict__ elog, const float* __restrict__ sbuf,
                          const float* __restrict__ xl, float* __restrict__ acc,
                          int nE, int nN) {
  int wid  = (int)(((size_t)blockIdx.x * blockDim.x + threadIdx.x) >> 5);
  int lane = threadIdx.x & 31;
  int tot = nE + nN;
  if (wid >= tot) return;
  int s, d;
  if (wid < nE) { s = src[wid]; d = dst[wid]; }
  else          { s = d = wid - nE; }
  const float* xs = xl + (size_t)s * HC;
  float* ad = acc + (size_t)d * HC;
#pragma unroll
  for (int h = 0; h < HEADS; ++h) {
    float alpha = elog[(size_t)wid * HEADS + h] / sbuf[(size_t)d * HEADS + h];
    int c = h * 32 + lane;
    atomicAdd(&ad[c], alpha * xs[c]);
  }
}

// ---------------- BN column statistics (coalesced: thread == column) ----------------
__global__ void __launch_bounds__(128) k_colstats(const float* __restrict__ x,
                                                  float* __restrict__ gsum,
                                                  float* __restrict__ gsq, int n) {
  int c = threadIdx.x;                         // 0..127
  int chunk = (n + gridDim.x - 1) / gridDim.x;
  int r0 = blockIdx.x * chunk;
  int r1 = min(r0 + chunk, n);
  float s = 0.f, q = 0.f;
  for (int r = r0; r < r1; ++r) {
    float v = x[(size_t)r * HC + c];
    s += v; q += v * v;
  }
  atomicAdd(&gsum[c], s);
  atomicAdd(&gsq[c], q);
}

// ---------------- BN apply + ELU  (bias1 omitted: cancels inside x - mean(x)) ----------------
__global__ void k_bnapply(const float* __restrict__ x, const float* __restrict__ gsum,
                          const float* __restrict__ gsq, const float* __restrict__ gamma,
                          const float* __restrict__ beta, float* __restrict__ h1, int n) {
  size_t i = (size_t)blockIdx.x * blockDim.x + threadIdx.x;
  if (i >= (size_t)n * HC) return;
  int c = (int)(i & (HC - 1));
  float inv_n = 1.0f / (float)n;
  float mu  = gsum[c] * inv_n;
  float var = gsq[c] * inv_n - mu * mu;        // population var (ddof=0)
  float y = gamma[c] * (x[i] - mu) * rsqrtf(var + BN_EPS) + beta[c];
  h1[i] = y > 0.f ? y : __expf(y) - 1.f;       // ELU(alpha=1)
}

// ---------------- final fuse: (emb + h1 + (acc2 + bias2)) / 3 ----------------
__global__ void k_final(const float* __restrict__ emb, const float* __restrict__ h1,
                        const float* __restrict__ acc, const float* __restrict__ bias2,
                        float* __restrict__ out, int n) {
  size_t i = (size_t)blockIdx.x * blockDim.x + threadIdx.x;
  if (i >= (size_t)n * HC) return;
  int c = (int)(i & (HC - 1));
  out[i] = (emb[i] + h1[i] + acc[i] + bias2[c]) * (1.0f / 3.0f);
}

// ------------------------------------------------------------------
static inline unsigned nblk(size_t n, int b) { return (unsigned)((n + b - 1) / b); }

extern "C" void kernel_launch(void* const* d_in, const int* in_sizes, int n_in,
                              void* d_out, int out_size, void* d_ws, size_t ws_size,
                              hipStream_t stream) {
  (void)n_in; (void)out_size; (void)ws_size;
  const float* emb   = (const float*)d_in[0];
  const int*   eidx  = (const int*)  d_in[1];
  const float* ew    = (const float*)d_in[2];
  const float* Wl1   = (const float*)d_in[3];
  const float* bl1   = (const float*)d_in[4];
  const float* Wr1   = (const float*)d_in[5];
  const float* br1   = (const float*)d_in[6];
  const float* We1   = (const float*)d_in[7];
  const float* att1  = (const float*)d_in[8];
  /* bias1 (d_in[9]) cancels under BatchNorm mean subtraction */
  const float* gamma1= (const float*)d_in[10];
  const float* beta1 = (const float*)d_in[11];
  const float* Wl2   = (const float*)d_in[12];
  const float* bl2   = (const float*)d_in[13];
  const float* Wr2   = (const float*)d_in[14];
  const float* br2   = (const float*)d_in[15];
  const float* We2   = (const float*)d_in[16];
  const float* att2  = (const float*)d_in[17];
  const float* bias2 = (const float*)d_in[18];
  float* out = (float*)d_out;

  const int n   = in_sizes[0] / HC;   // 50000 nodes
  const int nE  = in_sizes[1] / 2;    // 800000 edges
  const int nE2 = nE + n;             // +self loops
  const int* src = eidx;
  const int* dst = eidx + nE;

  // workspace carve-up (floats)
  float* ws = (float*)d_ws;
  size_t off = 0;
  float*    loopat = ws + off; off += n;
  float*    deg    = ws + off; off += n;
  float*    sumw   = ws + off; off += n;
  float*    xl     = ws + off; off += (size_t)n * HC;
  float*    xr     = ws + off; off += (size_t)n * HC;
  float*    elog   = ws + off; off += (size_t)nE2 * HEADS;
  unsigned* mbuf   = (unsigned*)(ws + off); off += (size_t)n * HEADS;
  float*    sbuf   = ws + off; off += (size_t)n * HEADS;
  float*    acc    = ws + off; off += (size_t)n * HC;
  float*    h1     = ws + off; off += (size_t)n * HC;
  float*    gsum   = ws + off; off += HC;
  float*    gsq    = ws + off; off += HC;

  const int B = 256;
  const unsigned gemmBlocks  = nblk((size_t)(n / 16) * 8 * 32, B); // waves*32 threads
  const unsigned edgeWaveBlk = nblk((size_t)nE2 * 32, B);

  // ---- self-loop attrs (fill_value='mean') ----
  k_fill_f32<<<nblk(2 * (size_t)n, B), B, 0, stream>>>(deg, 0.f, 2 * (size_t)n); // deg+sumw contiguous
  k_degsum<<<nblk(nE, B), B, 0, stream>>>(dst, ew, deg, sumw, nE);
  k_loopattr<<<nblk(n, B), B, 0, stream>>>(deg, sumw, loopat, n);

  // ==== layer 1 ====
  k_gemm_wmma<<<gemmBlocks, B, 0, stream>>>(emb, Wl1, bl1, xl, n);
  k_gemm_wmma<<<gemmBlocks, B, 0, stream>>>(emb, Wr1, br1, xr, n);
  k_fill_u32<<<nblk((size_t)n * HEADS, B), B, 0, stream>>>(mbuf, NEG_INF_FLIPPED, (size_t)n * HEADS);
  k_fill_f32<<<nblk((size_t)n * HEADS, B), B, 0, stream>>>(sbuf, 0.f, (size_t)n * HEADS);
  k_fill_f32<<<nblk((size_t)n * HC, B), B, 0, stream>>>(acc, 0.f, (size_t)n * HC);
  k_logits <<<edgeWaveBlk, B, 0, stream>>>(src, dst, ew, loopat, xl, xr, We1, att1, elog, mbuf, nE, n);
  k_expsum <<<nblk((size_t)nE2 * HEADS, B), B, 0, stream>>>(dst, elog, mbuf, sbuf, nE, n);
  k_scatter<<<edgeWaveBlk, B, 0, stream>>>(src, dst, elog, sbuf, xl, acc, nE, n);
  // BN + ELU -> h1
  k_fill_f32<<<1, 2 * HC, 0, stream>>>(gsum, 0.f, 2 * HC);  // gsum+gsq contiguous
  k_colstats<<<250, HC, 0, stream>>>(acc, gsum, gsq, n);
  k_bnapply<<<nblk((size_t)n * HC, B), B, 0, stream>>>(acc, gsum, gsq, gamma1, beta1, h1, n);

  // ==== layer 2 ====
  k_gemm_wmma<<<gemmBlocks, B, 0, stream>>>(h1, Wl2, bl2, xl, n);
  k_gemm_wmma<<<gemmBlocks, B, 0, stream>>>(h1, Wr2, br2, xr, n);
  k_fill_u32<<<nblk((size_t)n * HEADS, B), B, 0, stream>>>(mbuf, NEG_INF_FLIPPED, (size_t)n * HEADS);
  k_fill_f32<<<nblk((size_t)n * HEADS, B), B, 0, stream>>>(sbuf, 0.f, (size_t)n * HEADS);
  k_fill_f32<<<nblk((size_t)n * HC, B), B, 0, stream>>>(acc, 0.f, (size_t)n * HC);
  k_logits <<<edgeWaveBlk, B, 0, stream>>>(src, dst, ew, loopat, xl, xr, We2, att2, elog, mbuf, nE, n);
  k_expsum <<<nblk((size_t)nE2 * HEADS, B), B, 0, stream>>>(dst, elog, mbuf, sbuf, nE, n);
  k_scatter<<<edgeWaveBlk, B, 0, stream>>>(src, dst, elog, sbuf, xl, acc, nE, n);

  // ==== fuse ====
  k_final<<<nblk((size_t)n * HC, B), B, 0, stream>>>(emb, h1, acc, bias2, out, n);
}